// InducingLocationsSpatialTransform_85323820303216
// MI455X (gfx1250) — compile-verified
//
#include <hip/hip_runtime.h>
#include <hip/hip_bf16.h>

// CDNA5 (gfx1250) spatial-transformer kernel, v2.
//   Phase 1: affine grid transform  Gs = A(2x3) @ Gt(3xHW)  via V_WMMA_F32_16X16X4_F32
//            (8 images x 16 pixels per WMMA, f32). The lane that owns each
//            (image,pixel) result finishes ALL per-pixel scalar work there
//            (floor/clamp/weights/corner byte-offsets) so phase 2 does none of it
//            redundantly across the 32 lanes. Staged in LDS as float4 + int4 (b128).
//   Phase 2: bilinear gather, lane = channel (wave32 == 32 channels): per pixel just
//            2 broadcast ds_load_b128, 4 v_add, 4 coalesced 128B global_load_b32,
//            mul+3fma, 1 coalesced store.

typedef float v2f __attribute__((ext_vector_type(2)));
typedef float v8f __attribute__((ext_vector_type(8)));

#define H  64
#define W  64
#define C  32
#define HW (H * W)
#define PIX_PER_BLOCK  256
#define IMGS_PER_BLOCK 8

__global__ __launch_bounds__(256)
void st_wmma_bilinear(const float* __restrict__ X,
                      const float* __restrict__ theta,
                      float* __restrict__ out) {
  // Per (img, pixLocal): bilinear weights (wa,wb,wc,wd) and 4 corner byte-offsets.
  __shared__ float4 wbuf[IMGS_PER_BLOCK * PIX_PER_BLOCK];   // 32 KB
  __shared__ int4   obuf[IMGS_PER_BLOCK * PIX_PER_BLOCK];   // 32 KB

  const int tid  = threadIdx.x;
  const int wave = tid >> 5;
  const int lane = tid & 31;
  const int half = lane >> 4;   // K pair select (A: K=0,1 vs K=2,3)
  const int m    = lane & 15;   // A row / B column index

  const int chunk   = blockIdx.x;                // pixel chunk: 0..HW/256-1
  const int group   = blockIdx.y;                // image group: 0..N/8-1
  const int pixBase = chunk * PIX_PER_BLOCK;

  // ---------------- Phase 1: coordinates via WMMA + per-pixel scalarized math ----
  // A-matrix (16x4), row M = 2*imgLocal + i  (i=0 -> x-row of affine, i=1 -> y-row).
  // Layout: lanes 0-15 hold K=0 (VGPR0) / K=1 (VGPR1); lanes 16-31 hold K=2 / K=3.
  {
    const int imgL = m >> 1;
    const int i    = m & 1;
    const int n    = group * IMGS_PER_BLOCK + imgL;
    const float* th = theta + n * 6 + i * 3;   // [a0, a1, a2] for this affine row

    v2f a;
    a.x = half ? th[2] : th[0];   // K=2 -> translation,  K=0 -> x-coeff
    a.y = half ? 0.0f  : th[1];   // K=3 -> zero,         K=1 -> y-coeff

    // 8 waves x 2 tiles x 16 pixels = 256 pixels per block
    #pragma unroll
    for (int t = 0; t < 2; ++t) {
      const int pixL = (wave * 2 + t) * 16 + m;         // local pixel 0..255
      const int pix  = pixBase + pixL;                  // global pixel in image
      const int col  = pix & (W - 1);
      const int row  = pix >> 6;
      const float xt = -1.0f + (2.0f / (float)(W - 1)) * (float)col;
      const float yt = -1.0f + (2.0f / (float)(H - 1)) * (float)row;

      // B-matrix (4x16): row K striped across lanes within a VGPR.
      // lanes 0-15: K=0 (xt) / K=1 (yt);  lanes 16-31: K=2 (ones) / K=3 (zero).
      v2f b;
      b.x = half ? 1.0f : xt;
      b.y = half ? 0.0f : yt;

      v8f c = {};
      v8f d = __builtin_amdgcn_wmma_f32_16x16x4_f32(
          /*neg_a=*/false, a, /*neg_b=*/false, b,
          /*c_mod=*/(short)0, c, /*reuse_a=*/false, /*reuse_b=*/false);

      // D[M][Npix]: VGPR r -> M=r (lanes 0-15), M=8+r (lanes 16-31).
      // M = 2*img + i, so (d[2j], d[2j+1]) = (Gs_x, Gs_y) for img j (+4 on hi half).
      #pragma unroll
      for (int j = 0; j < 4; ++j) {
        const int imgD = j + half * 4;
        // gx = W*(v+1)/2, gy = H*(v+1)/2  (H == W == 64)
        const float gx = 32.0f * (d[2 * j]     + 1.0f);
        const float gy = 32.0f * (d[2 * j + 1] + 1.0f);

        const int fx = (int)floorf(gx);
        const int fy = (int)floorf(gy);
        const int x0 = min(max(fx,     0), W - 1);
        const int x1 = min(max(fx + 1, 0), W - 1);
        const int y0 = min(max(fy,     0), H - 1);
        const int y1 = min(max(fy + 1, 0), H - 1);

        // weights from CLIPPED corner indices (exact reference semantics)
        const float wx1 = (float)x1 - gx;
        const float wx0 = gx - (float)x0;
        const float wy1 = (float)y1 - gy;
        const float wy0 = gy - (float)y0;

        const int idx = imgD * PIX_PER_BLOCK + pixL;
        wbuf[idx] = make_float4(wx1 * wy1,   // (Y0, X0)
                                wx1 * wy0,   // (Y1, X0)
                                wx0 * wy1,   // (Y0, X1)
                                wx0 * wy0);  // (Y1, X1)
        obuf[idx] = make_int4(((y0 * W + x0) * C) << 2,
                              ((y1 * W + x0) * C) << 2,
                              ((y0 * W + x1) * C) << 2,
                              ((y1 * W + x1) * C) << 2);
      }
    }
  }
  __syncthreads();

  // ---------------- Phase 2: bilinear gather, lane = channel ----------------
  const int img = wave;                                  // one image per wave
  const int n   = group * IMGS_PER_BLOCK + img;
  const char* __restrict__ Xb = (const char*)(X + (size_t)n * HW * C);
  float* __restrict__ Tn = out + ((size_t)n * HW + pixBase) * C;
  const int laneOff = lane << 2;

  #pragma unroll 4
  for (int p = 0; p < PIX_PER_BLOCK; ++p) {
    const float4 wv = wbuf[img * PIX_PER_BLOCK + p];     // broadcast ds_load_b128
    const int4   ov = obuf[img * PIX_PER_BLOCK + p];     // broadcast ds_load_b128

    // 4 coalesced 128B corner reads (lane = channel)
    const float v00 = *(const float*)(Xb + (ov.x + laneOff));
    const float v10 = *(const float*)(Xb + (ov.y + laneOff));
    const float v01 = *(const float*)(Xb + (ov.z + laneOff));
    const float v11 = *(const float*)(Xb + (ov.w + laneOff));

    Tn[p * C + lane] = wv.x * v00 + wv.y * v10 + wv.z * v01 + wv.w * v11;
  }
}

extern "C" void kernel_launch(void* const* d_in, const int* in_sizes, int n_in,
                              void* d_out, int out_size, void* d_ws, size_t ws_size,
                              hipStream_t stream) {
  const float* X     = (const float*)d_in[0];   // (N, 64, 64, 32) f32
  const float* theta = (const float*)d_in[1];   // (N, 6) f32
  float* out = (float*)d_out;                   // (N, 64, 64, 32) f32

  const int N = in_sizes[1] / 6;                // 512
  dim3 grid(HW / PIX_PER_BLOCK, N / IMGS_PER_BLOCK);  // (16, 64)
  st_wmma_bilinear<<<grid, dim3(256), 0, stream>>>(X, theta, out);
}